// RegionLoss_16406775070970
// MI455X (gfx1250) — compile-verified
//
#include <hip/hip_runtime.h>
#include <hip/hip_bf16.h>

#define HW 65536
#define KRANK 26213u          // int(65536*0.4 - 1)
#define NBATCH 128
#define CHUNKS 8              // blocks per batch; 8*256*32 px = 65536
#define ITERS 8               // per-thread float4 iterations (8 * 4 px = 32 px)

typedef __attribute__((ext_vector_type(2))) float v2f;
typedef __attribute__((ext_vector_type(8))) float v8f;

__device__ __forceinline__ float grayOf(float r, float g, float b) {
    // match left-to-right, non-contracted evaluation in every pass
    return __fadd_rn(__fadd_rn(__fmul_rn(0.39f, r), __fmul_rn(0.5f, g)),
                     __fmul_rn(0.11f, b));
}

__device__ __forceinline__ unsigned sortKey(float f) {
    unsigned x = __float_as_uint(f);
    return (x & 0x80000000u) ? ~x : (x | 0x80000000u);
}

// ---------------- Level-1 histogram: bits [31:21], 2048 bins ----------------
__global__ __launch_bounds__(256) void hist_l1(const float* __restrict__ yhat,
                                               unsigned* __restrict__ hist1) {
    __shared__ unsigned lh[2048];
    const int b = blockIdx.y, chunk = blockIdx.x, tid = threadIdx.x;
    for (int i = tid; i < 2048; i += 256) lh[i] = 0;
    __syncthreads();

    const float4* R  = (const float4*)(yhat + ((size_t)b * 3 + 0) * HW);
    const float4* G  = (const float4*)(yhat + ((size_t)b * 3 + 1) * HW);
    const float4* Bc = (const float4*)(yhat + ((size_t)b * 3 + 2) * HW);
    const int base4 = chunk * (HW / CHUNKS / 4);
    for (int it = 0; it < ITERS; ++it) {
        int i4 = base4 + it * 256 + tid;
        float4 r = R[i4], g = G[i4], bb = Bc[i4];
        atomicAdd(&lh[sortKey(grayOf(r.x, g.x, bb.x)) >> 21], 1u);
        atomicAdd(&lh[sortKey(grayOf(r.y, g.y, bb.y)) >> 21], 1u);
        atomicAdd(&lh[sortKey(grayOf(r.z, g.z, bb.z)) >> 21], 1u);
        atomicAdd(&lh[sortKey(grayOf(r.w, g.w, bb.w)) >> 21], 1u);
    }
    __syncthreads();
    for (int i = tid; i < 2048; i += 256) {
        unsigned v = lh[i];
        if (v) atomicAdd(&hist1[i * NBATCH + b], v);   // bin-major -> coalesced scan
    }
}

// ---------------- Level-2 histogram: bits [20:10] within selected l1 bin ----
__global__ __launch_bounds__(256) void hist_l2(const float* __restrict__ yhat,
                                               const unsigned* __restrict__ bin1,
                                               unsigned* __restrict__ hist2) {
    __shared__ unsigned lh[2048];
    const int b = blockIdx.y, chunk = blockIdx.x, tid = threadIdx.x;
    const unsigned b1 = bin1[b];
    for (int i = tid; i < 2048; i += 256) lh[i] = 0;
    __syncthreads();

    const float4* R  = (const float4*)(yhat + ((size_t)b * 3 + 0) * HW);
    const float4* G  = (const float4*)(yhat + ((size_t)b * 3 + 1) * HW);
    const float4* Bc = (const float4*)(yhat + ((size_t)b * 3 + 2) * HW);
    const int base4 = chunk * (HW / CHUNKS / 4);
    for (int it = 0; it < ITERS; ++it) {
        int i4 = base4 + it * 256 + tid;
        float4 r = R[i4], g = G[i4], bb = Bc[i4];
        unsigned u;
        u = sortKey(grayOf(r.x, g.x, bb.x)); if ((u >> 21) == b1) atomicAdd(&lh[(u >> 10) & 2047u], 1u);
        u = sortKey(grayOf(r.y, g.y, bb.y)); if ((u >> 21) == b1) atomicAdd(&lh[(u >> 10) & 2047u], 1u);
        u = sortKey(grayOf(r.z, g.z, bb.z)); if ((u >> 21) == b1) atomicAdd(&lh[(u >> 10) & 2047u], 1u);
        u = sortKey(grayOf(r.w, g.w, bb.w)); if ((u >> 21) == b1) atomicAdd(&lh[(u >> 10) & 2047u], 1u);
    }
    __syncthreads();
    for (int i = tid; i < 2048; i += 256) {
        unsigned v = lh[i];
        if (v) atomicAdd(&hist2[i * NBATCH + b], v);
    }
}

// ---------------- Level-3 histogram: bits [9:0], 1024 bins ------------------
__global__ __launch_bounds__(256) void hist_l3(const float* __restrict__ yhat,
                                               const unsigned* __restrict__ bin1,
                                               const unsigned* __restrict__ bin2,
                                               unsigned* __restrict__ hist3) {
    __shared__ unsigned lh[1024];
    const int b = blockIdx.y, chunk = blockIdx.x, tid = threadIdx.x;
    const unsigned hi = (bin1[b] << 11) | bin2[b];   // bits [31:10]
    for (int i = tid; i < 1024; i += 256) lh[i] = 0;
    __syncthreads();

    const float4* R  = (const float4*)(yhat + ((size_t)b * 3 + 0) * HW);
    const float4* G  = (const float4*)(yhat + ((size_t)b * 3 + 1) * HW);
    const float4* Bc = (const float4*)(yhat + ((size_t)b * 3 + 2) * HW);
    const int base4 = chunk * (HW / CHUNKS / 4);
    for (int it = 0; it < ITERS; ++it) {
        int i4 = base4 + it * 256 + tid;
        float4 r = R[i4], g = G[i4], bb = Bc[i4];
        unsigned u;
        u = sortKey(grayOf(r.x, g.x, bb.x)); if ((u >> 10) == hi) atomicAdd(&lh[u & 1023u], 1u);
        u = sortKey(grayOf(r.y, g.y, bb.y)); if ((u >> 10) == hi) atomicAdd(&lh[u & 1023u], 1u);
        u = sortKey(grayOf(r.z, g.z, bb.z)); if ((u >> 10) == hi) atomicAdd(&lh[u & 1023u], 1u);
        u = sortKey(grayOf(r.w, g.w, bb.w)); if ((u >> 10) == hi) atomicAdd(&lh[u & 1023u], 1u);
    }
    __syncthreads();
    for (int i = tid; i < 1024; i += 256) {
        unsigned v = lh[i];
        if (v) atomicAdd(&hist3[i * NBATCH + b], v);
    }
}

// ------------- Scan a bin-major histogram; pick bin containing target -------
__global__ void scan_bins(const unsigned* __restrict__ hist, int nbins,
                          const unsigned* __restrict__ remIn,
                          unsigned* __restrict__ binOut,
                          unsigned* __restrict__ remOut) {
    const int b = threadIdx.x;                 // 128 threads, one per batch
    unsigned target = remIn ? remIn[b] : KRANK;
    unsigned c = 0, sel = (unsigned)nbins - 1u, below = 0;
    for (int bin = 0; bin < nbins; ++bin) {
        unsigned c2 = c + hist[bin * NBATCH + b];
        if (target < c2) { sel = (unsigned)bin; below = c; break; }
        c = c2;
    }
    binOut[b] = sel;
    remOut[b] = target - below;
}

// ------------- Final scan: reconstruct exact k-th value, yu = -kth ----------
__global__ void scan_final(const unsigned* __restrict__ hist3,
                           const unsigned* __restrict__ bin1,
                           const unsigned* __restrict__ bin2,
                           const unsigned* __restrict__ rem2,
                           float* __restrict__ yuF) {
    const int b = threadIdx.x;
    unsigned target = rem2[b];
    unsigned c = 0, sel = 1023u;
    for (int bin = 0; bin < 1024; ++bin) {
        unsigned c2 = c + hist3[bin * NBATCH + b];
        if (target < c2) { sel = (unsigned)bin; break; }
        c = c2;
    }
    unsigned u = (bin1[b] << 21) | (bin2[b] << 10) | sel;
    unsigned bits = (u & 0x80000000u) ? (u ^ 0x80000000u) : ~u;
    yuF[b] = -__uint_as_float(bits);
}

// ------------- Wave (32-lane) fp32 sum via V_WMMA_F32_16X16X4_F32 -----------
// A(16x4): lane L<16 -> A[L, {0,1}] = {s, 0}; lane L>=16 -> A[L-16, {2,3}].
// B(4x16) = all ones => D[m,n] = s_m + s_{m+16}. Lane 0 holds rows 0..7 in
// its 8 accumulators, lane 16 holds rows 8..15 -> two shuffles finish.
__device__ __forceinline__ float wave_sum(float s) {
#if __has_builtin(__builtin_amdgcn_wmma_f32_16x16x4_f32)
    v2f a; a[0] = s;    a[1] = 0.0f;
    v2f o; o[0] = 1.0f; o[1] = 1.0f;
    v8f c = {};
    c = __builtin_amdgcn_wmma_f32_16x16x4_f32(false, a, false, o,
                                              (short)0, c, false, false);
    float t = ((c[0] + c[1]) + (c[2] + c[3])) + ((c[4] + c[5]) + (c[6] + c[7]));
    return __shfl(t, 0, 32) + __shfl(t, 16, 32);
#else
    for (int off = 16; off; off >>= 1) s += __shfl_xor(s, off, 32);
    return s;
#endif
}

// ------------- Final pass: weighted L1 over both tensors --------------------
__global__ __launch_bounds__(256) void final_loss(const float* __restrict__ yhat,
                                                  const float* __restrict__ y,
                                                  const float* __restrict__ yuF,
                                                  float* __restrict__ partials) {
    const int b = blockIdx.y, chunk = blockIdx.x, tid = threadIdx.x;
    const float yu = yuF[b];

    const float4* R  = (const float4*)(yhat + ((size_t)b * 3 + 0) * HW);
    const float4* G  = (const float4*)(yhat + ((size_t)b * 3 + 1) * HW);
    const float4* Bc = (const float4*)(yhat + ((size_t)b * 3 + 2) * HW);
    const float4* Ry = (const float4*)(y + ((size_t)b * 3 + 0) * HW);
    const float4* Gy = (const float4*)(y + ((size_t)b * 3 + 1) * HW);
    const float4* By = (const float4*)(y + ((size_t)b * 3 + 2) * HW);

    const int base4 = chunk * (HW / CHUNKS / 4);
    float acc = 0.0f;
    for (int it = 0; it < ITERS; ++it) {
        int i4 = base4 + it * 256 + tid;
        __builtin_prefetch(&Ry[i4 + 256], 0, 0);   // global_prefetch_b8
        float4 r = R[i4], g = G[i4], bb = Bc[i4];
        float4 r2 = Ry[i4], g2 = Gy[i4], b2 = By[i4];
        {
            float w = (grayOf(r.x, g.x, bb.x) <= yu) ? 4.0f : 1.0f;
            acc += w * (fabsf(r.x - r2.x) + fabsf(g.x - g2.x) + fabsf(bb.x - b2.x));
        }
        {
            float w = (grayOf(r.y, g.y, bb.y) <= yu) ? 4.0f : 1.0f;
            acc += w * (fabsf(r.y - r2.y) + fabsf(g.y - g2.y) + fabsf(bb.y - b2.y));
        }
        {
            float w = (grayOf(r.z, g.z, bb.z) <= yu) ? 4.0f : 1.0f;
            acc += w * (fabsf(r.z - r2.z) + fabsf(g.z - g2.z) + fabsf(bb.z - b2.z));
        }
        {
            float w = (grayOf(r.w, g.w, bb.w) <= yu) ? 4.0f : 1.0f;
            acc += w * (fabsf(r.w - r2.w) + fabsf(g.w - g2.w) + fabsf(bb.w - b2.w));
        }
    }

    // EXEC is all-ones here (no divergence at block granularity) -> WMMA legal
    float wsum = wave_sum(acc);

    __shared__ float wpart[8];
    const int lane = tid & 31, wv = tid >> 5;
    if (lane == 0) wpart[wv] = wsum;
    __syncthreads();
    if (tid == 0) {
        float t = 0.0f;
        for (int i = 0; i < 8; ++i) t += wpart[i];   // fixed order
        partials[b * CHUNKS + chunk] = t;
    }
}

// ------------- Deterministic final reduction of 1024 partials ---------------
__global__ void reduce_final(const float* __restrict__ partials, float* __restrict__ out) {
    __shared__ float sh[256];
    const int tid = threadIdx.x;
    float s = 0.0f;
    for (int i = tid; i < NBATCH * CHUNKS; i += 256) s += partials[i];  // fixed order
    sh[tid] = s;
    __syncthreads();
    if (tid == 0) {
        float t = 0.0f;
        for (int i = 0; i < 256; ++i) t += sh[i];    // fixed order
        out[0] = t * (1.0f / 25165824.0f);           // / (128*3*65536)
    }
}

extern "C" void kernel_launch(void* const* d_in, const int* in_sizes, int n_in,
                              void* d_out, int out_size, void* d_ws, size_t ws_size,
                              hipStream_t stream) {
    const float* yhat = (const float*)d_in[0];
    const float* y    = (const float*)d_in[1];

    char* ws = (char*)d_ws;
    unsigned* hist1 = (unsigned*)(ws);                     // 2048*128*4 = 1 MB
    unsigned* hist2 = (unsigned*)(ws + 1048576);           // 1 MB
    unsigned* hist3 = (unsigned*)(ws + 2097152);           // 0.5 MB
    unsigned* bin1  = (unsigned*)(ws + 2621440);           // 128
    unsigned* rem1  = bin1 + 128;
    unsigned* bin2  = bin1 + 256;
    unsigned* rem2  = bin1 + 384;
    float*    yuF   = (float*)(bin1 + 512);
    float*    parts = (float*)(bin1 + 640);                // 1024 floats

    // zero the three histogram regions (contiguous 2.5 MB)
    hipMemsetAsync(ws, 0, 2621440, stream);

    dim3 grid(CHUNKS, NBATCH), blk(256);
    hist_l1<<<grid, blk, 0, stream>>>(yhat, hist1);
    scan_bins<<<1, NBATCH, 0, stream>>>(hist1, 2048, nullptr, bin1, rem1);
    hist_l2<<<grid, blk, 0, stream>>>(yhat, bin1, hist2);
    scan_bins<<<1, NBATCH, 0, stream>>>(hist2, 2048, rem1, bin2, rem2);
    hist_l3<<<grid, blk, 0, stream>>>(yhat, bin1, bin2, hist3);
    scan_final<<<1, NBATCH, 0, stream>>>(hist3, bin1, bin2, rem2, yuF);
    final_loss<<<grid, blk, 0, stream>>>(yhat, y, yuF, parts);
    reduce_final<<<1, 256, 0, stream>>>(parts, (float*)d_out);
}